// GATDotConv_45578192945955
// MI455X (gfx1250) — compile-verified
//
#include <hip/hip_runtime.h>
#include <float.h>

// Problem constants (from reference): N nodes, E edges, F_IN features,
// H heads, D head-dim, F_OUT = H*D.
#define NN    50000
#define NE    800000
#define FIN   128
#define NH    4
#define DD    32
#define FOUT  128   // NH * DD

typedef __attribute__((ext_vector_type(2))) float v2f;
typedef __attribute__((ext_vector_type(8))) float v8f;

// ---------------------------------------------------------------------------
// Float atomic max via signed/unsigned integer ordering trick.
// Requires the cell to be initialized with -FLT_MAX (bit pattern works for
// both orderings). Correct for all finite mixed-sign values.
// ---------------------------------------------------------------------------
__device__ __forceinline__ void atomicMaxFloat(float* addr, float val) {
    if (!(val < 0.0f)) {
        atomicMax((int*)addr, __float_as_int(val));
    } else {
        atomicMin((unsigned int*)addr, __float_as_uint(val));
    }
}

// ---------------------------------------------------------------------------
// Kernel 0: initialize output with bias, emax with -FLT_MAX, denom with 0.
// ---------------------------------------------------------------------------
__global__ void gat_init_kernel(float* __restrict__ out,
                                const float* __restrict__ bias,
                                float* __restrict__ emax,
                                float* __restrict__ denom) {
    int gid = blockIdx.x * blockDim.x + threadIdx.x;
    if (gid < NN * FOUT) out[gid] = bias[gid & (FOUT - 1)];
    if (gid < NN * NH) { emax[gid] = -FLT_MAX; denom[gid] = 0.0f; }
}

// ---------------------------------------------------------------------------
// Kernel 1: ft = feat @ W  via V_WMMA_F32_16X16X4_F32 (exact fp32).
// Block = 256 threads = 8 waves. Wave w of block b computes the 16x16 output
// tile (rows b*16..b*16+15, cols w*16..w*16+15). K loop: 128/4 = 32 WMMAs.
//
// fp32 A 16x4 layout: lanes 0-15 hold M=lane, {VGPR0,VGPR1} = K{0,1};
//                     lanes 16-31 hold M=lane-16, K{2,3}.
// fp32 B 4x16 layout mirrors A (row striped across lanes within a VGPR).
// fp32 C/D 16x16: VGPR r = row (r + 8*half), col = lane&15.
// ---------------------------------------------------------------------------
__global__ void gat_gemm_wmma_kernel(const float* __restrict__ A,   // [NN, FIN]
                                     const float* __restrict__ Wm,  // [FIN, FOUT]
                                     float* __restrict__ ft) {      // [NN, FOUT]
    const int wave = threadIdx.x >> 5;    // 0..7 -> tileN
    const int lane = threadIdx.x & 31;
    const int half = lane >> 4;           // 0: K pair {0,1}, 1: K pair {2,3}
    const int l    = lane & 15;

    const int row = blockIdx.x * 16 + l;  // A row handled by this lane
    const int col = wave * 16 + l;        // B col handled by this lane

    const float* __restrict__ arow = A + row * FIN;

    v8f acc = {};
    #pragma unroll
    for (int k0 = 0; k0 < FIN; k0 += 4) {
        const int ka = k0 + 2 * half;
        v2f a, b;
        a.x = arow[ka];
        a.y = arow[ka + 1];
        b.x = Wm[ka * FOUT + col];
        b.y = Wm[(ka + 1) * FOUT + col];
        acc = __builtin_amdgcn_wmma_f32_16x16x4_f32(
            /*neg_a=*/false, a, /*neg_b=*/false, b,
            /*c_mod=*/(short)0, acc, /*reuse_a=*/false, /*reuse_b=*/false);
    }

    float* __restrict__ orow = ft + (blockIdx.x * 16 + 8 * half) * FOUT + wave * 16 + l;
    #pragma unroll
    for (int r = 0; r < 8; ++r) {
        orow[r * FOUT] = acc[r];
    }
}

// ---------------------------------------------------------------------------
// Kernel 2: per-(edge,head) dot product e[e,h] = <ft[src,h,:], ft[dst,h,:]>
// and atomic segment-max into emax[dst,h]. ft (25.6 MB) is L2-resident.
// ---------------------------------------------------------------------------
__global__ void gat_edge_dot_kernel(const float* __restrict__ ft,
                                    const int* __restrict__ src,
                                    const int* __restrict__ dst,
                                    float* __restrict__ e,
                                    float* __restrict__ emax) {
    int gid = blockIdx.x * blockDim.x + threadIdx.x;   // [0, NE*NH)
    if (gid >= NE * NH) return;
    const int edge = gid >> 2;      // / NH
    const int h    = gid & 3;       // % NH
    const int s = src[edge];
    const int d = dst[edge];

    const float4* __restrict__ ps = (const float4*)(ft + s * FOUT + h * DD);
    const float4* __restrict__ pd = (const float4*)(ft + d * FOUT + h * DD);
    float acc = 0.0f;
    #pragma unroll
    for (int i = 0; i < DD / 4; ++i) {
        float4 a = ps[i];
        float4 b = pd[i];
        acc += a.x * b.x + a.y * b.y + a.z * b.z + a.w * b.w;
    }
    e[gid] = acc;
    atomicMaxFloat(&emax[d * NH + h], acc);
}

// ---------------------------------------------------------------------------
// Kernel 3: ex = exp(e - emax[dst]); accumulate denom[dst] (segment sum).
// ex overwrites e in place.
// ---------------------------------------------------------------------------
__global__ void gat_edge_exp_kernel(const int* __restrict__ dst,
                                    float* __restrict__ e,
                                    const float* __restrict__ emax,
                                    float* __restrict__ denom) {
    int gid = blockIdx.x * blockDim.x + threadIdx.x;
    if (gid >= NE * NH) return;
    const int edge = gid >> 2;
    const int h    = gid & 3;
    const int d    = dst[edge];
    const float ex = __expf(e[gid] - emax[d * NH + h]);
    e[gid] = ex;
    atomicAdd(&denom[d * NH + h], ex);
}

// ---------------------------------------------------------------------------
// Kernel 4: weighted scatter:  out[dst,h,:] += a * ft[src,h,:],
// a = ex / denom[dst,h]. One wave (32 lanes) per edge; lane = head-dim index
// -> each head's 32 atomics form one fully coalesced 128B burst.
// ---------------------------------------------------------------------------
__global__ void gat_scatter_kernel(const float* __restrict__ ft,
                                   const int* __restrict__ src,
                                   const int* __restrict__ dst,
                                   const float* __restrict__ ex,
                                   const float* __restrict__ denom,
                                   float* __restrict__ out) {
    const int lane = threadIdx.x & 31;
    const int edge = blockIdx.x * (blockDim.x >> 5) + (threadIdx.x >> 5);
    if (edge >= NE) return;
    const int s = src[edge];
    const int d = dst[edge];

    #pragma unroll
    for (int h = 0; h < NH; ++h) {
        const float a = ex[edge * NH + h] / denom[d * NH + h];
        const float v = ft[s * FOUT + h * DD + lane] * a;
        atomicAdd(&out[d * FOUT + h * DD + lane], v);
    }
}

// ---------------------------------------------------------------------------
// Launch. Inputs (setup_inputs order): feat[N,128] f32, src[E] int, dst[E] int,
// W[128,128] f32, bias[128] f32. Output: [N,H,D] f32 (row-major = [N,128]).
// Workspace layout (floats): ft[NN*FOUT] | e[NE*NH] | emax[NN*NH] | denom[NN*NH]
// -> ~40.1 MB total.
// ---------------------------------------------------------------------------
extern "C" void kernel_launch(void* const* d_in, const int* in_sizes, int n_in,
                              void* d_out, int out_size, void* d_ws, size_t ws_size,
                              hipStream_t stream) {
    const float* feat = (const float*)d_in[0];
    const int*   src  = (const int*)d_in[1];
    const int*   dst  = (const int*)d_in[2];
    const float* Wm   = (const float*)d_in[3];
    const float* bias = (const float*)d_in[4];
    float* out = (float*)d_out;

    float* ft    = (float*)d_ws;
    float* e     = ft + (size_t)NN * FOUT;
    float* emax  = e  + (size_t)NE * NH;
    float* denom = emax + (size_t)NN * NH;

    // 0: init out = bias, emax = -inf-ish, denom = 0
    {
        int threads = NN * FOUT;                       // 6.4M (covers NN*NH too)
        int blocks = (threads + 255) / 256;
        gat_init_kernel<<<blocks, 256, 0, stream>>>(out, bias, emax, denom);
    }
    // 1: ft = feat @ W  (WMMA fp32)
    {
        gat_gemm_wmma_kernel<<<NN / 16, 256, 0, stream>>>(feat, Wm, ft);
    }
    // 2: edge dots + segment max
    {
        int threads = NE * NH;                         // 3.2M
        int blocks = (threads + 255) / 256;
        gat_edge_dot_kernel<<<blocks, 256, 0, stream>>>(ft, src, dst, e, emax);
    }
    // 3: exp + segment sum
    {
        int threads = NE * NH;
        int blocks = (threads + 255) / 256;
        gat_edge_exp_kernel<<<blocks, 256, 0, stream>>>(dst, e, emax, denom);
    }
    // 4: weighted scatter (wave per edge)
    {
        int edges_per_block = 256 / 32;                // 8
        int blocks = (NE + edges_per_block - 1) / edges_per_block;  // 100000
        gat_scatter_kernel<<<blocks, 256, 0, stream>>>(ft, src, dst, e, denom, out);
    }
}